// MLPDecoder_68092411511098
// MI455X (gfx1250) — compile-verified
//
#include <hip/hip_runtime.h>

typedef __attribute__((ext_vector_type(2))) float v2f;
typedef __attribute__((ext_vector_type(4))) float v4f;
typedef __attribute__((ext_vector_type(8))) float v8f;

#define LD   16
#define NROW 2048

// ---------------------------------------------------------------------------
// Stage 1: hi = z @ W1[:16], hjb = z @ W1[16:] + b1   (both 2048x16, fp32)
// One wave per 16-row tile of z; K=16 via 4 chained V_WMMA_F32_16X16X4_F32.
// ---------------------------------------------------------------------------
__global__ __launch_bounds__(32) void mlp_stage1(const float* __restrict__ z,
                                                 const float* __restrict__ W1,
                                                 const float* __restrict__ b1,
                                                 float* __restrict__ hi,
                                                 float* __restrict__ hjb) {
  const int lane = threadIdx.x;       // 0..31
  const int r    = lane & 15;         // row within tile (A), col (B/D)
  const int half = lane >> 4;         // 0: K0/K1, 1: K2/K3
  const int tile = blockIdx.x;        // 0..127
  const int row  = tile * 16 + r;

  v8f acc_i = {};
  v8f acc_j = {};
#pragma unroll
  for (int kk = 0; kk < 4; ++kk) {
    const int k0 = kk * 4 + half * 2;
    // A (16x4 f32): lanes 0-15 -> K = k0, k0+1 ; lanes 16-31 -> K = k0+2, k0+3
    v2f a;
    a.x = z[row * LD + k0];
    a.y = z[row * LD + k0 + 1];
    // B (4x16 f32): VGPR0 = rows K0/K2 striped over lane halves, VGPR1 = K1/K3
    v2f bi, bj;
    bi.x = W1[(k0    ) * LD + r];         // W1_i = W1[:16]
    bi.y = W1[(k0 + 1) * LD + r];
    bj.x = W1[(16 + k0    ) * LD + r];    // W1_j = W1[16:]
    bj.y = W1[(16 + k0 + 1) * LD + r];
    acc_i = __builtin_amdgcn_wmma_f32_16x16x4_f32(false, a, false, bi, (short)0,
                                                  acc_i, false, false);
    acc_j = __builtin_amdgcn_wmma_f32_16x16x4_f32(false, a, false, bj, (short)0,
                                                  acc_j, false, false);
  }
  const float bv = b1[r];
#pragma unroll
  for (int v = 0; v < 8; ++v) {
    // D layout: VGPR v holds M=v (lanes 0-15) and M=v+8 (lanes 16-31), N=r
    const int m = tile * 16 + v + half * 8;
    hi [m * LD + r] = acc_i[v];
    hjb[m * LD + r] = acc_j[v] + bv;
  }
}

// ---------------------------------------------------------------------------
// Stage 2: out[i,j] = sigmoid(b2 + sum_k w2[k] * relu(hi[i,k] + hjb[j,k]))
// One wave per 16x16 output tile. Per k, the broadcast-add matrix
// hi[i,k]+hjb[j,k] is rank-2 -> one V_WMMA_F32_16X16X4_F32 with
// A = [hi[:,k] | 1], B = [1 ; hjb[:,k]^T] (K=2,3 lanes zeroed).
// ---------------------------------------------------------------------------
__global__ __launch_bounds__(256) void mlp_stage2(const float* __restrict__ hi,
                                                  const float* __restrict__ hjb,
                                                  const float* __restrict__ W2,
                                                  const float* __restrict__ b2,
                                                  float* __restrict__ out) {
  const int lane   = threadIdx.x & 31;
  const int wave   = threadIdx.x >> 5;
  const int tileId = blockIdx.x * 8 + wave;     // 0..16383
  const int ti     = tileId >> 7;               // 0..127 (i tile)
  const int tj     = tileId & 127;              // 0..127 (j tile)
  const int r      = lane & 15;
  const int half   = lane >> 4;
  const bool lo    = (half == 0);

  // Transposed tiles: lane L (<16) holds row L of hi/hjb tile (16 floats).
  v4f ha[4] = {{0,0,0,0},{0,0,0,0},{0,0,0,0},{0,0,0,0}};
  v4f hb[4] = {{0,0,0,0},{0,0,0,0},{0,0,0,0},{0,0,0,0}};
  if (lo) {
    const v4f* pa = (const v4f*)(hi  + (ti * 16 + lane) * LD);
    const v4f* pb = (const v4f*)(hjb + (tj * 16 + lane) * LD);
#pragma unroll
    for (int w = 0; w < 4; ++w) { ha[w] = pa[w]; hb[w] = pb[w]; }
  }
  const float one_lo = lo ? 1.0f : 0.0f;

  // Uniform scalars (compiler promotes to s_loads).
  float w2[LD];
#pragma unroll
  for (int k = 0; k < LD; ++k) w2[k] = W2[k];
  const float bias = b2[0];

  v8f acc = {bias, bias, bias, bias, bias, bias, bias, bias};
#pragma unroll
  for (int k = 0; k < LD; ++k) {
    v2f a, b;
    a.x = ha[k >> 2][k & 3];   // A[m,0] = hi[i,k]   (0 in lanes 16-31)
    a.y = one_lo;              // A[m,1] = 1         (0 in lanes 16-31 -> K3)
    b.x = one_lo;              // B[0,n] = 1
    b.y = hb[k >> 2][k & 3];   // B[1,n] = hjb[j,k]
    v8f zero = {};
    v8f d = __builtin_amdgcn_wmma_f32_16x16x4_f32(false, a, false, b, (short)0,
                                                  zero, false, false);
#pragma unroll
    for (int v = 0; v < 8; ++v)
      acc[v] += fmaxf(d[v], 0.0f) * w2[k];
  }

#pragma unroll
  for (int v = 0; v < 8; ++v) {
    const float x = acc[v];
    const float e = __builtin_amdgcn_exp2f(x * -1.44269504088896340736f); // exp(-x)
    const float s = __builtin_amdgcn_rcpf(1.0f + e);                      // sigmoid
    out[(ti * 16 + v + half * 8) * NROW + tj * 16 + r] = s;
  }
}

// ---------------------------------------------------------------------------
extern "C" void kernel_launch(void* const* d_in, const int* in_sizes, int n_in,
                              void* d_out, int out_size, void* d_ws, size_t ws_size,
                              hipStream_t stream) {
  const float* z  = (const float*)d_in[0];   // 2048 x 16
  const float* W1 = (const float*)d_in[1];   // 32 x 16
  const float* b1 = (const float*)d_in[2];   // 16
  const float* W2 = (const float*)d_in[3];   // 16 x 1
  const float* b2 = (const float*)d_in[4];   // 1
  float* out = (float*)d_out;                // 2048 x 2048

  float* hi  = (float*)d_ws;                 // 2048 x 16
  float* hjb = hi + NROW * LD;               // 2048 x 16

  mlp_stage1<<<NROW / 16, 32, 0, stream>>>(z, W1, b1, hi, hjb);
  mlp_stage2<<<(NROW / 16) * (NROW / 16) / 8, 256, 0, stream>>>(hi, hjb, W2, b2, out);
}